// SparseSlotAttentionSelector_39960375722244
// MI455X (gfx1250) — compile-verified
//
#include <hip/hip_runtime.h>
#include <hip/hip_bf16.h>

// ---------------------------------------------------------------------------
// SparseSlotAttentionSelector on MI455X (gfx1250)
//
// Memory-bound (201 MB in @ 23.3 TB/s ~= 8.6 us floor, ~1 GFLOP): single pass
// over local_feats, fp32 end-to-end. Logits GEMM on V_WMMA_F32_16X16X4_F32.
// x tiles are streamed HBM -> LDS with GLOBAL_LOAD_ASYNC_TO_LDS_B128
// (ASYNCcnt-tracked), double-buffered per wave so chunk k+1 is in flight
// while WMMAs consume chunk k. All intermediates live in LDS.
// ---------------------------------------------------------------------------

typedef __attribute__((ext_vector_type(2))) float v2f;
typedef __attribute__((ext_vector_type(8))) float v8f;

#define B_      64
#define N_      1024
#define D_      768
#define K_      8
#define PPS_    8
#define THREADS 512
#define WAVES   16
#define TEMP_   0.1f
#define NEGINF_ -10000.0f
#define EPS_    1e-6f

// LDS layout in floats:
//   s_lds : 16*768 slots, rows 8..15 zero           [0, 12288)
//   rs    : 16 recip slot norms                     [12288, 12304)
//   thr   : 16 top-8 thresholds                     [12304, 12320)
//   part  : 64 gram partials                        [12320, 12384)
//   logit : 8*1024 logits (reused in-place as attn) [12384, 20576)
//   xtile : WAVES * 2 * (16patch x 64dim) dbl-buf   [20576, 53344)  (16B aligned)
#define OFF_S     0
#define OFF_RS    12288
#define OFF_THR   12304
#define OFF_PART  12320
#define OFF_LOGIT 12384
#define OFF_XT    20576
#define LDS_FLOATS (OFF_XT + WAVES * 2 * 16 * 64)   // 53344 floats = 213376 B < 320 KB

// Issue 8x b128 async copies: one 16-patch x 64-dim fp32 chunk -> LDS buffer.
// s_wait_dscnt 0 first: guarantees all prior ds_loads of the buffer being
// overwritten have fully completed (async-unit LDS writes bypass the wave's
// in-order DS pipe, so WAR must be closed explicitly).
__device__ __forceinline__ void issue_chunk_async(const float* __restrict__ gsrc,
                                                  float* __restrict__ ldst,
                                                  int lane)
{
    asm volatile("s_wait_dscnt 0x0" ::: "memory");
    #pragma unroll
    for (int i = 0; i < 8; ++i) {
        const int q  = i * 32 + lane;   // 256 float4 total
        const int r  = q >> 4;          // patch row (16 lanes span one 64f row)
        const int cq = q & 15;          // float4 column
        // VDST = LDS byte offset (low 32 bits of the flat shared pointer)
        const unsigned lds_off =
            (unsigned)(uintptr_t)(ldst + r * 64 + cq * 4);
        const unsigned long long gaddr =
            (unsigned long long)(uintptr_t)(gsrc + (size_t)r * D_ + cq * 4);
        asm volatile("global_load_async_to_lds_b128 %0, %1, off"
                     :: "v"(lds_off), "v"(gaddr)
                     : "memory");
    }
}

__global__ void slot_attn_main(const float* __restrict__ lf,
                               const float* __restrict__ slots,
                               float* __restrict__ out,
                               float* __restrict__ ws)
{
    extern __shared__ float lds[];
    float* s_lds = lds + OFF_S;
    float* rs    = lds + OFF_RS;
    float* thr   = lds + OFF_THR;
    float* part  = lds + OFF_PART;
    float* logit = lds + OFF_LOGIT;
    float* xtile = lds + OFF_XT;

    const int b    = blockIdx.x;
    const int tid  = threadIdx.x;
    const int lane = tid & 31;     // wave32
    const int wave = tid >> 5;
    const int l16  = lane & 15;
    const int half = lane >> 4;    // selects K-pair per ISA 16x4 f32 A-layout

    // ---- stage 0: slots -> LDS (zero-pad rows 8..15), reciprocal norms ----
    for (int i = tid; i < 16 * D_; i += THREADS) {
        const int r = i / D_;
        s_lds[i] = (r < K_) ? slots[i] : 0.0f;
    }
    __syncthreads();

    if (wave < K_) {
        const float* row = s_lds + wave * D_;
        float ss = 0.0f;
        for (int i = lane; i < D_; i += 32) { const float v = row[i]; ss += v * v; }
        for (int off = 16; off > 0; off >>= 1) ss += __shfl_xor(ss, off, 32);
        if (lane == 0) rs[wave] = 1.0f / fmaxf(sqrtf(ss), EPS_);
    }
    __syncthreads();

    // ---- stage 1: logits via WMMA, async double-buffered x streaming ----
    const float* xb_base = lf + (size_t)b * N_ * D_;
    float* xt0 = xtile + wave * (2 * 16 * 64);
    float* xt1 = xt0 + 16 * 64;

    for (int t = wave; t < N_ / 16; t += WAVES) {
        const int n0 = t * 16;
        const float* xb = xb_base + (size_t)n0 * D_;

        v8f acc = {0.f, 0.f, 0.f, 0.f, 0.f, 0.f, 0.f, 0.f};
        float xx = 0.0f;

        // prefetch chunk 0
        issue_chunk_async(xb, xt0, lane);

        for (int c = 0; c < D_ / 64; ++c) {
            const int k0 = c * 64;
            float* cur = (c & 1) ? xt1 : xt0;

            if (c + 1 < D_ / 64) {
                // start fetching chunk c+1 into the other buffer, then wait
                // for chunk c only (async loads complete in order: <=8
                // outstanding means the first 8 have landed).
                issue_chunk_async(xb + k0 + 64, (c & 1) ? xt0 : xt1, lane);
                asm volatile("s_wait_asynccnt 0x8" ::: "memory");
            } else {
                asm volatile("s_wait_asynccnt 0x0" ::: "memory");
            }

            // 16x WMMA F32 16x16x4: D += A(slots 16x4) x B(x-chunk 4x16)
            #pragma unroll
            for (int kk = 0; kk < 64; kk += 4) {
                v2f a, bm;
                const int ka = k0 + kk + 2 * half;   // A: M=l16, K pair by half
                a.x = s_lds[l16 * D_ + ka];
                a.y = s_lds[l16 * D_ + ka + 1];
                const int kb = kk + 2 * half;        // B: N=l16, K pair by half
                bm.x = cur[l16 * 64 + kb];
                bm.y = cur[l16 * 64 + kb + 1];
                acc = __builtin_amdgcn_wmma_f32_16x16x4_f32(
                    false, a, false, bm, (short)0, acc, false, false);
            }

            // x.x partial: lane covers half the dims of patch l16
            const float* xr = cur + l16 * 64 + 32 * half;
            #pragma unroll
            for (int dd = 0; dd < 32; ++dd) { const float v = xr[dd]; xx += v * v; }
        }

        // combine the two half-dim partials of patch l16 (lanes L <-> L^16)
        xx += __shfl_xor(xx, 16, 32);
        const float rxv = 1.0f / fmaxf(sqrtf(xx), EPS_);

        // C/D layout: comp r, lanes 0..15 -> (M=r, N=lane); slots 8..15 zero pad
        if (lane < 16) {
            #pragma unroll
            for (int r = 0; r < K_; ++r)
                logit[r * N_ + n0 + lane] = acc[r] * rs[r] * rxv;
        }
    }
    __syncthreads();

    // ---- stage 2: per-row top-8 threshold (iterative strict-max exclusion) ----
    if (wave < K_) {
        const float* row = logit + wave * N_;
        float cur = 3.4e38f;
        for (int it = 0; it < PPS_; ++it) {
            float m = -3.4e38f;
            for (int i = lane; i < N_; i += 32) {
                const float v = row[i];
                m = (v < cur) ? fmaxf(m, v) : m;
            }
            for (int off = 16; off > 0; off >>= 1) m = fmaxf(m, __shfl_xor(m, off, 32));
            cur = m;
        }
        if (lane == 0) thr[wave] = cur;
    }
    __syncthreads();

    // ---- stage 3: img_space_mask + slots_b broadcast (reads logits before reuse) ----
    float* out_mask = out + (size_t)B_ * K_ * D_ + 1;
    for (int n = tid; n < N_; n += THREADS) {
        float mmax = -3.4e38f;
        #pragma unroll
        for (int kk = 0; kk < K_; ++kk) {
            const float l = logit[kk * N_ + n];
            const float h = (l >= thr[kk]) ? 1.0f : 0.0f;
            const float s = 1.0f / (1.0f + __expf(-l * (1.0f / TEMP_)));
            const float m = (h - s) + s;   // matches reference fp order
            mmax = fmaxf(mmax, m);
        }
        out_mask[(size_t)b * N_ + n] = mmax;
    }
    for (int i = tid; i < K_ * D_; i += THREADS)
        out[(size_t)b * K_ * D_ + i] = slots[i];
    __syncthreads();

    // ---- stage 4: masked softmax per row, in place (waves 0..7) ----
    if (wave < K_) {
        float* row = logit + wave * N_;
        const float tk = thr[wave];
        float mmax = -3.4e38f;
        for (int i = lane; i < N_; i += 32) {
            const float l = row[i];
            const float h = (l >= tk) ? 1.0f : 0.0f;
            const float s = 1.0f / (1.0f + __expf(-l * (1.0f / TEMP_)));
            const float m = (h - s) + s;
            const float ml = l * m + NEGINF_ * (1.0f - m);
            row[i] = ml;
            mmax = fmaxf(mmax, ml);
        }
        for (int off = 16; off > 0; off >>= 1) mmax = fmaxf(mmax, __shfl_xor(mmax, off, 32));
        float sum = 0.0f;
        for (int i = lane; i < N_; i += 32) {
            const float e = __expf(row[i] - mmax);
            row[i] = e;
            sum += e;
        }
        for (int off = 16; off > 0; off >>= 1) sum += __shfl_xor(sum, off, 32);
        const float inv = 1.0f / sum;
        for (int i = lane; i < N_; i += 32) row[i] *= inv;
    }
    __syncthreads();

    // ---- stage 5: 8x8 gram + |G - I| partial (4 (k,j) pairs per wave) ----
    {
        const int p0 = wave * 4;
        #pragma unroll
        for (int pi = 0; pi < 4; ++pi) {
            const int p  = p0 + pi;
            const int kk = p >> 3, jj = p & 7;
            const float* ra = logit + kk * N_;
            const float* rb = logit + jj * N_;
            float dot = 0.0f;
            for (int i = lane; i < N_; i += 32) dot += ra[i] * rb[i];
            for (int off = 16; off > 0; off >>= 1) dot += __shfl_xor(dot, off, 32);
            if (lane == 0) part[p] = fabsf(dot - ((kk == jj) ? 1.0f : 0.0f));
        }
    }
    __syncthreads();

    if (wave == 0) {
        float s = part[lane] + part[lane + 32];
        for (int off = 16; off > 0; off >>= 1) s += __shfl_xor(s, off, 32);
        if (lane == 0) ws[b] = s;
    }
}

// Deterministic final reduce of the 64 per-batch partials (fixed shuffle order).
__global__ void slot_attn_finish(const float* __restrict__ ws, float* __restrict__ out)
{
    const int lane = threadIdx.x;
    float s = ws[lane] + ws[lane + 32];
    for (int off = 16; off > 0; off >>= 1) s += __shfl_xor(s, off, 32);
    if (lane == 0)
        out[(size_t)B_ * K_ * D_] = s * (1.0f / (float)(B_ * K_ * K_));
}

extern "C" void kernel_launch(void* const* d_in, const int* in_sizes, int n_in,
                              void* d_out, int out_size, void* d_ws, size_t ws_size,
                              hipStream_t stream)
{
    const float* lf    = (const float*)d_in[0];   // (64, 1024, 768) f32
    const float* slots = (const float*)d_in[1];   // (1, 8, 768) f32
    float* out = (float*)d_out;                   // slots_b | ortho_loss | img_space_mask
    float* ws  = (float*)d_ws;                    // 64 f32 partials

    const size_t shmem = (size_t)LDS_FLOATS * sizeof(float);
    slot_attn_main<<<B_, THREADS, shmem, stream>>>(lf, slots, out, ws);
    slot_attn_finish<<<1, 32, 0, stream>>>(ws, out);
}